// PeriodicVolumeEncoding_8538394984936
// MI455X (gfx1250) — compile-verified
//
#include <hip/hip_runtime.h>
#include <cmath>
#include <cstdint>

#define NUM_LEVELS 16
#define L2_HASH 18           // log2 hashmap size
#define PVR_MASK 63          // PVR = 2^(18//3) = 64

struct Scalings { float s[NUM_LEVELS]; };

// ---- CDNA5 async global->LDS path (guarded: compiles either way) ----------
#if __has_builtin(__builtin_amdgcn_global_load_async_to_lds_b32)
#define HAVE_ASYNC_LDS 1
typedef __attribute__((address_space(1))) int as1_int;
typedef __attribute__((address_space(3))) int as3_int;
#else
#define HAVE_ASYNC_LDS 0
#endif

#if HAVE_ASYNC_LDS
__device__ __forceinline__ void wait_asynccnt0() {
#if __has_builtin(__builtin_amdgcn_s_wait_asynccnt)
  __builtin_amdgcn_s_wait_asynccnt(0);
#else
  asm volatile("s_wait_asynccnt 0" ::: "memory");
#endif
}
#endif

// One thread per (point, level). Lanes 0..15 of a wave: same point, levels 0..15.
// Block = 256 threads = 16 points x 16 levels.
__global__ __launch_bounds__(256) void pve_hashgrid_kernel(
    const float*  __restrict__ in_tensor,   // [N,3]
    const float2* __restrict__ table,       // [16*2^18, 2] as float2
    double*       __restrict__ out,         // [N*16] float2 punned to double
    Scalings sc, int n_points)
{
  const int tid = threadIdx.x;
  const int t   = blockIdx.x * 256 + tid;
  const int n_threads = n_points * NUM_LEVELS;

  __shared__ float s_sc[NUM_LEVELS];
#if HAVE_ASYNC_LDS
  __shared__ float pts[16 * 3];
  {
    const int g = blockIdx.x * 48 + tid;        // 16 points * 3 coords per block
    if (tid < 48 && g < n_points * 3) {
      __builtin_amdgcn_global_load_async_to_lds_b32(
          (as1_int*)(uintptr_t)(in_tensor + g),
          (as3_int*)(uint32_t)(uintptr_t)(&pts[tid]),
          0, 0);
    }
  }
#endif
  if (tid < NUM_LEVELS) s_sc[tid] = sc.s[tid];
#if HAVE_ASYNC_LDS
  wait_asynccnt0();
#endif
  __syncthreads();

  if (t >= n_threads) return;
  const int point = t >> 4;
  const int level = t & 15;

#if HAVE_ASYNC_LDS
  const int lp = tid >> 4;
  const float px = pts[lp * 3 + 0];
  const float py = pts[lp * 3 + 1];
  const float pz = pts[lp * 3 + 2];
#else
  const float px = in_tensor[point * 3 + 0];
  const float py = in_tensor[point * 3 + 1];
  const float pz = in_tensor[point * 3 + 2];
#endif

  const float s  = s_sc[level];
  const float sx = px * s, sy = py * s, sz = pz * s;
  const float fxf = floorf(sx), fyf = floorf(sy), fzf = floorf(sz);
  const float ox = sx - fxf, oy = sy - fyf, oz = sz - fzf;
  const int fx = (int)fxf,      fy = (int)fyf,      fz = (int)fzf;
  const int cx = (int)ceilf(sx), cy = (int)ceilf(sy), cz = (int)ceilf(sz);

  // hash = (x%64)*4096 + (y%64)*64 + (z%64) + level*2^18  (coords >= 0)
  const int base = level << L2_HASH;
  const int xc = (cx & PVR_MASK) << 12, xf = (fx & PVR_MASK) << 12;
  const int yc = (cy & PVR_MASK) << 6,  yf = (fy & PVR_MASK) << 6;
  const int zc = (cz & PVR_MASK),       zf = (fz & PVR_MASK);

  // corner order matches _CORNER_IS_CEIL rows (x,y,z ceil flags):
  // 0:(1,1,1) 1:(1,0,1) 2:(0,0,1) 3:(0,1,1) 4:(1,1,0) 5:(1,0,0) 6:(0,0,0) 7:(0,1,0)
  const float2 f0 = table[base + xc + yc + zc];
  const float2 f1 = table[base + xc + yf + zc];
  const float2 f2 = table[base + xf + yf + zc];
  const float2 f3 = table[base + xf + yc + zc];
  const float2 f4 = table[base + xc + yc + zf];
  const float2 f5 = table[base + xc + yf + zf];
  const float2 f6 = table[base + xf + yf + zf];
  const float2 f7 = table[base + xf + yc + zf];

  const float rx = 1.0f - ox, ry = 1.0f - oy, rz = 1.0f - oz;
  const float a03x = f0.x * ox + f3.x * rx, a03y = f0.y * ox + f3.y * rx;
  const float a12x = f1.x * ox + f2.x * rx, a12y = f1.y * ox + f2.y * rx;
  const float a56x = f5.x * ox + f6.x * rx, a56y = f5.y * ox + f6.y * rx;
  const float a47x = f4.x * ox + f7.x * rx, a47y = f4.y * ox + f7.y * rx;
  const float bx = a03x * oy + a12x * ry,   by = a03y * oy + a12y * ry;
  const float gx = a47x * oy + a56x * ry,   gy = a47y * oy + a56y * ry;

  float2 enc;
  enc.x = bx * oz + gx * rz;
  enc.y = by * oz + gy * rz;

  // Streaming output (written once, never re-read): non-temporal so the 32 MB
  // result stream does not evict the L2-resident 32 MB hash table.
  __builtin_nontemporal_store(__builtin_bit_cast(double, enc), out + t);
}

extern "C" void kernel_launch(void* const* d_in, const int* in_sizes, int n_in,
                              void* d_out, int out_size, void* d_ws, size_t ws_size,
                              hipStream_t stream) {
  (void)n_in; (void)out_size; (void)d_ws; (void)ws_size;
  const float*  in_tensor = (const float*)d_in[0];
  const float2* table     = (const float2*)d_in[1];
  const int n_points = in_sizes[0] / 3;

  // Replicate numpy's float64 chain for SCALINGS (resolves the near-integer
  // floor cases at levels 5/10/15 the same way libm does).
  Scalings sc;
  const double growth = std::exp((std::log(1024.0) - std::log(16.0)) / 15.0);
  for (int l = 0; l < NUM_LEVELS; ++l)
    sc.s[l] = (float)std::floor(16.0 * std::pow(growth, (double)l));

  const int n_threads = n_points * NUM_LEVELS;
  const int blocks = (n_threads + 255) / 256;
  pve_hashgrid_kernel<<<blocks, 256, 0, stream>>>(in_tensor, table,
                                                  (double*)d_out, sc, n_points);
}